// SampleGNN_52329881534940
// MI455X (gfx1250) — compile-verified
//
#include <hip/hip_runtime.h>
#include <hip/hip_bf16.h>

typedef float v2f __attribute__((ext_vector_type(2)));
typedef float v8f __attribute__((ext_vector_type(8)));

#define BLK_M 128
#define KT 64
#define AL_STRIDE 68   // pad 64->68: conflict-free b64 LDS reads, 16B-aligned rows

// ---------------------------------------------------------------------------
// fp32 WMMA GEMM:  C[M x 16*NT] = act( scale_row(A[M x K]) @ B[K x 16*NT] + bias )
// BLK_M=128 rows per block, 8 waves, wave w owns rows [w*16, w*16+16), all cols.
// K must be a multiple of KT=64. NT (cols/16) is a compile-time template arg so
// the inner loop is straight-line WMMA with no scalar branches.
// Double-buffered LDS: next K-slice is prefetched into registers while the
// current slice is consumed by v_wmma_f32_16x16x4_f32; one barrier per slice.
// degv != nullptr applies per-row scale 1/max(degv[r],1e-8) (GCN normalization).
// ---------------------------------------------------------------------------
template <int NT>
__global__ __launch_bounds__(256) void wmma_gemm_kernel(
    const float* __restrict__ A, const float* __restrict__ degv,
    const float* __restrict__ B, const float* __restrict__ bias,
    float* __restrict__ C, int M, int K, int doRelu)
{
    constexpr int NC   = NT * 16;            // output columns
    constexpr int BLSZ = (KT / 4) * NC * 4;  // swizzled B slice, floats

    __shared__ float Al[2][BLK_M * AL_STRIDE];
    __shared__ float Bl[2][BLSZ];
    __shared__ float rscale[BLK_M];

    const int tid  = threadIdx.x;
    const int wave = tid >> 5;
    const int lane = tid & 31;
    const int l16  = lane & 15;
    const int hi2  = (lane >= 16) ? 2 : 0;   // lanes 16..31 carry K+2/K+3 halves
    const long long row0 = (long long)blockIdx.x * BLK_M;

    if (tid < BLK_M) {
        long long r = row0 + tid;
        float s = 1.0f;
        if (degv != nullptr && r < M) s = 1.0f / fmaxf(degv[r], 1e-8f);
        rscale[tid] = s;
    }
    __syncthreads();

    // ---- register prefetch state: A tile 8 float4, B tile NT float4 ----
    float4 av[8];
    float4 bv[NT];

    auto loadTile = [&](int kb) {
#pragma unroll
        for (int it = 0; it < 8; ++it) {
            int f  = tid + it * 256;
            int r  = f >> 4;
            int c4 = (f & 15) << 2;
            long long row = row0 + r;
            av[it] = make_float4(0.f, 0.f, 0.f, 0.f);
            if (row < M) av[it] = *(const float4*)(A + row * (long long)K + kb + c4);
        }
#pragma unroll
        for (int it = 0; it < NT; ++it) {
            int f  = tid + it * 256;
            int k  = f / (NC / 4);
            int c4 = (f - k * (NC / 4)) << 2;
            bv[it] = *(const float4*)(B + (long long)(kb + k) * NC + c4);
        }
    };

    auto storeTile = [&](int p) {
#pragma unroll
        for (int it = 0; it < 8; ++it) {
            int f  = tid + it * 256;
            int r  = f >> 4;
            int c4 = (f & 15) << 2;
            float s  = rscale[r];
            float* d = &Al[p][r * AL_STRIDE + c4];
            d[0] = av[it].x * s; d[1] = av[it].y * s;
            d[2] = av[it].z * s; d[3] = av[it].w * s;
        }
#pragma unroll
        for (int it = 0; it < NT; ++it) {
            int f  = tid + it * 256;
            int k  = f / (NC / 4);
            int c4 = (f - k * (NC / 4)) << 2;
            float* d = &Bl[p][(k >> 2) * (NC * 4) + (k & 3)];
            d[(c4 + 0) * 4] = bv[it].x;
            d[(c4 + 1) * 4] = bv[it].y;
            d[(c4 + 2) * 4] = bv[it].z;
            d[(c4 + 3) * 4] = bv[it].w;
        }
    };

    v8f acc[NT];
#pragma unroll
    for (int i = 0; i < NT; ++i) { v8f z = {}; acc[i] = z; }

    const int arow = wave * 16 + l16;
    const int nkb  = K / KT;

    // prologue: tile 0 -> LDS buffer 0
    loadTile(0);
    storeTile(0);
    __syncthreads();

    int p = 0;
    for (int t = 0; t < nkb; ++t) {
        if (t + 1 < nkb) loadTile((t + 1) * KT);   // global loads in flight

        const float* Ap = &Al[p][arow * AL_STRIDE + hi2];
        const float* Bp = &Bl[p][(l16 << 2) + hi2];
#pragma unroll
        for (int k = 0; k < KT; k += 4) {
            // A fragment 16x4: lanes 0-15 -> K{k,k+1}, lanes 16-31 -> K{k+2,k+3}
            v2f a = *(const v2f*)(Ap + k);
            v2f bfr[NT];
#pragma unroll
            for (int nt = 0; nt < NT; ++nt)
                bfr[nt] = *(const v2f*)(Bp + (k >> 2) * (NC * 4) + nt * 64);
#pragma unroll
            for (int nt = 0; nt < NT; ++nt)
                acc[nt] = __builtin_amdgcn_wmma_f32_16x16x4_f32(
                    false, a, false, bfr[nt], (short)0, acc[nt], false, false);
        }

        if (t + 1 < nkb) storeTile(p ^ 1);
        __syncthreads();
        p ^= 1;
    }

    // ---- epilogue: C/D layout -> VGPR v holds M=v (lanes 0-15) / M=8+v ----
#pragma unroll
    for (int nt = 0; nt < NT; ++nt) {
        int col = nt * 16 + l16;
        float bvv = (bias != nullptr) ? bias[col] : 0.0f;
        long long rbase = row0 + wave * 16 + (hi2 ? 8 : 0);
#pragma unroll
        for (int v = 0; v < 8; ++v) {
            long long row = rbase + v;
            if (row < M) {
                float val = acc[nt][v] + bvv;
                if (doRelu) val = fmaxf(val, 0.0f);
                C[row * (long long)NC + col] = val;
            }
        }
    }
}

// ---------------------------------------------------------------------------
__global__ void zero_kernel(float* __restrict__ p, long long n)
{
    long long i = (long long)blockIdx.x * blockDim.x + threadIdx.x;
    if (i < n) p[i] = 0.0f;
}

__global__ void degree_kernel(const int* __restrict__ dst, const float* __restrict__ w,
                              float* __restrict__ deg, int E)
{
    int e = blockIdx.x * blockDim.x + threadIdx.x;
    if (e < E) atomicAdd(&deg[dst[e]], w[e]);
}

// msg = h[src]*w scatter-added into hnew[dst]; one thread per (edge, 4 cols).
// h (51 MB) is L2-resident on the 192 MB L2, so the random gather is cheap.
__global__ void gather_scatter_kernel(const float* __restrict__ h,
                                      const int* __restrict__ src,
                                      const int* __restrict__ dst,
                                      const float* __restrict__ w,
                                      float* __restrict__ hnew,
                                      long long total, int D4, int D)
{
    long long i = (long long)blockIdx.x * blockDim.x + threadIdx.x;
    if (i >= total) return;
    int e  = (int)(i / D4);
    int c4 = (int)(i - (long long)e * D4) * 4;
    int s = src[e], d = dst[e];
    float wt = w[e];
    const float4 hv = *(const float4*)(h + (long long)s * D + c4);
    float* o = hnew + (long long)d * D + c4;
    atomicAdd(o + 0, hv.x * wt);
    atomicAdd(o + 1, hv.y * wt);
    atomicAdd(o + 2, hv.z * wt);
    atomicAdd(o + 3, hv.w * wt);
}

// Per-column sum / sumsq over the node axis (BatchNorm training stats).
#define STAT_ROWS 512
__global__ __launch_bounds__(256) void bn_stats_kernel(const float* __restrict__ z,
    float* __restrict__ stats, int Nn, int C)
{
    int tid = threadIdx.x;
    int c = tid % C;               // C is 64 or 128 -> divides 256
    int rofs = tid / C;
    int rstride = 256 / C;
    long long rbeg = (long long)blockIdx.x * STAT_ROWS;
    long long rend = rbeg + STAT_ROWS;
    if (rend > Nn) rend = Nn;
    float s = 0.0f, s2 = 0.0f;
    for (long long r = rbeg + rofs; r < rend; r += rstride) {
        float v = z[r * C + c];
        s += v; s2 += v * v;
    }
    atomicAdd(&stats[c], s);
    atomicAdd(&stats[C + c], s2);
}

// tgt = (addTo ? tgt : 0) + act((z - mu) * rsqrt(var+eps) * g + b)
__global__ void bn_apply_kernel(const float* __restrict__ z,
                                const float* __restrict__ stats,
                                const float* __restrict__ g,
                                const float* __restrict__ b,
                                float* __restrict__ tgt,
                                long long total, int C, float invN,
                                int addTo, int doRelu)
{
    long long i = (long long)blockIdx.x * blockDim.x + threadIdx.x;
    if (i >= total) return;
    int c = (int)(i % C);
    float mean = stats[c] * invN;
    float var  = stats[C + c] * invN - mean * mean;   // biased variance (jnp.var)
    float rs   = rsqrtf(var + 1e-5f);
    float v = (z[i] - mean) * rs * g[c] + b[c];
    if (doRelu) v = fmaxf(v, 0.0f);
    if (addTo) tgt[i] += v; else tgt[i] = v;
}

// Final tiny head: out[N x T] = z[N x H] @ W[H x T] + b  (0.1 GFLOP -> VALU)
__global__ __launch_bounds__(256) void head_out_kernel(const float* __restrict__ z,
    const float* __restrict__ W, const float* __restrict__ bias,
    float* __restrict__ out, int Nn, int H, int T)
{
    __shared__ float Wl[128 * 16];
    __shared__ float bl[16];
    for (int i = threadIdx.x; i < H * T; i += blockDim.x) Wl[i] = W[i];
    if (threadIdx.x < T) bl[threadIdx.x] = bias[threadIdx.x];
    __syncthreads();
    long long n = (long long)blockIdx.x * blockDim.x + threadIdx.x;
    if (n >= Nn) return;
    float accT[8];
#pragma unroll
    for (int t = 0; t < 8; ++t) accT[t] = (t < T) ? bl[t] : 0.0f;
    for (int k = 0; k < H; ++k) {
        float hv = z[n * (long long)H + k];
#pragma unroll
        for (int t = 0; t < 8; ++t)
            if (t < T) accT[t] += hv * Wl[k * T + t];
    }
    for (int t = 0; t < T && t < 8; ++t) out[n * (long long)T + t] = accT[t];
}

// ---------------------------------------------------------------------------
static inline unsigned cdivu(long long a, long long b) { return (unsigned)((a + b - 1) / b); }

extern "C" void kernel_launch(void* const* d_in, const int* in_sizes, int n_in,
                              void* d_out, int out_size, void* d_ws, size_t ws_size,
                              hipStream_t stream)
{
    const float* x     = (const float*)d_in[0];
    const int*   eidx  = (const int*)  d_in[1];
    const float* ew    = (const float*)d_in[2];
    const float* W_in  = (const float*)d_in[3];
    const float* b_in  = (const float*)d_in[4];
    const float* gcn_W = (const float*)d_in[5];
    const float* gcn_b = (const float*)d_in[6];
    const float* bn_g  = (const float*)d_in[7];
    const float* bn_b  = (const float*)d_in[8];
    const float* hW1   = (const float*)d_in[9];
    const float* hb1   = (const float*)d_in[10];
    const float* hbn_g = (const float*)d_in[11];
    const float* hbn_b = (const float*)d_in[12];
    const float* hW2   = (const float*)d_in[13];
    const float* hb2   = (const float*)d_in[14];
    float* out = (float*)d_out;

    const int D   = in_sizes[4];            // 128
    const int DIN = in_sizes[3] / D;        // 2048
    const int N   = in_sizes[0] / DIN;      // 100000
    const int E   = in_sizes[2];            // 1600000
    const int L   = in_sizes[6] / D;        // 2
    const int H   = in_sizes[10];           // 64
    const int T   = in_sizes[14];           // 8

    // ---- workspace layout ----
    char* ws = (char*)d_ws;
    auto al256 = [](size_t v) { return (v + 255) & ~(size_t)255; };
    size_t sA = al256((size_t)N * D * sizeof(float));
    float* hA    = (float*)(ws);                 // node features h
    float* hB    = (float*)(ws + sA);            // scatter accumulator / z1
    float* hC    = (float*)(ws + 2 * sA);        // GEMM output / head activations
    float* deg   = (float*)(ws + 3 * sA);
    float* stats = (float*)(ws + 3 * sA + al256((size_t)N * sizeof(float)));

    const int* srcv = eidx;
    const int* dstv = eidx + E;

    // degrees (layer-invariant)
    zero_kernel<<<cdivu(N, 256), 256, 0, stream>>>(deg, N);
    degree_kernel<<<cdivu(E, 256), 256, 0, stream>>>(dstv, ew, deg, E);

    // h = relu(x @ W_in + b_in)   -- the 52 GFLOP / 819 MB kernel
    wmma_gemm_kernel<8><<<cdivu(N, BLK_M), 256, 0, stream>>>(
        x, nullptr, W_in, b_in, hA, N, DIN, 1);

    for (int i = 0; i < L; ++i) {
        long long nd = (long long)N * D;
        zero_kernel<<<cdivu(nd, 256), 256, 0, stream>>>(hB, nd);
        long long tot = (long long)E * (D / 4);
        gather_scatter_kernel<<<cdivu(tot, 256), 256, 0, stream>>>(
            hA, srcv, dstv, ew, hB, tot, D / 4, D);
        zero_kernel<<<1, 256, 0, stream>>>(stats, 2 * D);
        // z = relu((hB/deg) @ W_i + b_i)
        wmma_gemm_kernel<8><<<cdivu(N, BLK_M), 256, 0, stream>>>(
            hB, deg, gcn_W + (size_t)i * D * D, gcn_b + (size_t)i * D, hC, N, D, 1);
        bn_stats_kernel<<<cdivu(N, STAT_ROWS), 256, 0, stream>>>(hC, stats, N, D);
        // h += BN(z)
        bn_apply_kernel<<<cdivu(nd, 256), 256, 0, stream>>>(
            hC, stats, bn_g + (size_t)i * D, bn_b + (size_t)i * D,
            hA, nd, D, 1.0f / (float)N, 1, 0);
    }

    // head: z1 = h @ hW1 + hb1 ; z = relu(BN(z1)) ; out = z @ hW2 + hb2
    zero_kernel<<<1, 256, 0, stream>>>(stats, 2 * H);
    wmma_gemm_kernel<4><<<cdivu(N, BLK_M), 256, 0, stream>>>(
        hA, nullptr, hW1, hb1, hB, N, D, 0);
    bn_stats_kernel<<<cdivu(N, STAT_ROWS), 256, 0, stream>>>(hB, stats, N, H);
    long long nh = (long long)N * H;
    bn_apply_kernel<<<cdivu(nh, 256), 256, 0, stream>>>(
        hB, stats, hbn_g, hbn_b, hC, nh, H, 1.0f / (float)N, 0, 1);
    head_out_kernel<<<cdivu(N, 256), 256, 0, stream>>>(hC, hW2, hb2, out, N, H, T);

    (void)n_in; (void)out_size; (void)ws_size;
}